// IntersectionalVolumeRatio_75883482185766
// MI455X (gfx1250) — compile-verified
//
#include <hip/hip_runtime.h>
#include <cstdint>

// Tile sizes
#define MT 32      // mentions per block
#define NT 64      // entities per block
#define DD 64      // box dims (128-wide embeds split into min/max halves of 64)
#define LSTR 68    // LDS row stride in floats: 64 data + 4 pad = 272B (17 x 16B)
                   // -> keeps every 16B chunk aligned, avoids pow2 bank pattern

// out[m, n] = prod_d  softplus(min(Zm,Ze) - max(zm,ze)) / softplus(Zm - zm)
// Each per-dim ratio is <= 1 (hi-lo <= Zm-zm, softplus monotone), so the direct
// product cannot overflow; underflow->0 matches exp(-large) of the reference.

__device__ __forceinline__ float softplus_f(float x) {
    // numerically stable softplus; accurate down to f32 underflow
    return fmaxf(x, 0.0f) + log1pf(__expf(-fabsf(x)));
}

__global__ __launch_bounds__(256) void IntersectionalVolumeRatio_kernel(
    const float* __restrict__ men, const float* __restrict__ en,
    float* __restrict__ out, int M, int N)
{
    __shared__ float zm_s[MT * LSTR];
    __shared__ float Zm_s[MT * LSTR];
    __shared__ float ze_s[NT * LSTR];
    __shared__ float Ze_s[NT * LSTR];
    __shared__ float ipm_s[MT * LSTR];   // 1 / softplus(Zm - zm)

    const int t      = threadIdx.x;
    const int n_base = blockIdx.x * NT;
    const int m_base = blockIdx.y * MT;

    // ---- Stage tiles into LDS with CDNA5 async global->LDS loads ----------
    // Mention tile: MT rows x 128 floats = MT*32 16-byte chunks.
    #pragma unroll
    for (int i = 0; i < (MT * 32) / 256; ++i) {
        int chunk = i * 256 + t;          // 0 .. MT*32-1
        int row   = chunk >> 5;           // chunk / 32
        int c     = chunk & 31;           // 16B chunk within the 512B source row
        if (m_base + row < M) {
            uint32_t goff = (uint32_t)(((m_base + row) * 128 + c * 4) * 4);
            float* dst = (c < 16) ? &zm_s[row * LSTR + c * 4]
                                  : &Zm_s[row * LSTR + (c - 16) * 4];
            uint32_t lds = (uint32_t)(uintptr_t)dst;   // low 32 bits = LDS byte addr
            asm volatile("global_load_async_to_lds_b128 %0, %1, %2"
                         :: "v"(lds), "v"(goff), "s"(men) : "memory");
        }
    }
    // Entity tile: NT rows x 128 floats = NT*32 chunks.
    #pragma unroll
    for (int i = 0; i < (NT * 32) / 256; ++i) {
        int chunk = i * 256 + t;
        int row   = chunk >> 5;
        int c     = chunk & 31;
        if (n_base + row < N) {
            uint32_t goff = (uint32_t)(((n_base + row) * 128 + c * 4) * 4);
            float* dst = (c < 16) ? &ze_s[row * LSTR + c * 4]
                                  : &Ze_s[row * LSTR + (c - 16) * 4];
            uint32_t lds = (uint32_t)(uintptr_t)dst;
            asm volatile("global_load_async_to_lds_b128 %0, %1, %2"
                         :: "v"(lds), "v"(goff), "s"(en) : "memory");
        }
    }
    // Drain this wave's ASYNCcnt, then workgroup barrier so all waves see LDS.
    asm volatile("s_wait_asynccnt 0x0" ::: "memory");
    __syncthreads();

    // ---- Precompute 1/softplus(Zm - zm) per (mention, dim) -----------------
    for (int i = t; i < MT * DD; i += 256) {
        int row = i >> 6, d = i & 63;
        if (m_base + row < M) {
            int idx = row * LSTR + d;
            ipm_s[idx] = 1.0f / softplus_f(Zm_s[idx] - zm_s[idx]);
        }
    }
    __syncthreads();

    // ---- Main compute: each thread = 1 entity column x 8 mentions ----------
    const int j  = t & (NT - 1);   // entity within tile (0..63)
    const int m0 = t >> 6;         // mention group (0..3), wave-uniform
                                   // -> zm/Zm/ipm b128 reads are LDS broadcasts

    float prod[8];
    #pragma unroll
    for (int k = 0; k < 8; ++k) prod[k] = 1.0f;

    #pragma unroll 2
    for (int dd = 0; dd < DD / 4; ++dd) {          // 4 dims per iteration, b128 LDS
        const float4 zev = *(const float4*)&ze_s[j * LSTR + dd * 4];
        const float4 Zev = *(const float4*)&Ze_s[j * LSTR + dd * 4];
        #pragma unroll
        for (int k = 0; k < 8; ++k) {
            const int mrow = (m0 + 4 * k) * LSTR + dd * 4;
            const float4 zmv = *(const float4*)&zm_s[mrow];
            const float4 Zmv = *(const float4*)&Zm_s[mrow];
            const float4 ipv = *(const float4*)&ipm_s[mrow];
            float p = prod[k];
            p *= softplus_f(fminf(Zmv.x, Zev.x) - fmaxf(zmv.x, zev.x)) * ipv.x;
            p *= softplus_f(fminf(Zmv.y, Zev.y) - fmaxf(zmv.y, zev.y)) * ipv.y;
            p *= softplus_f(fminf(Zmv.z, Zev.z) - fmaxf(zmv.z, zev.z)) * ipv.z;
            p *= softplus_f(fminf(Zmv.w, Zev.w) - fmaxf(zmv.w, zev.w)) * ipv.w;
            prod[k] = p;
        }
    }

    // ---- Store (lane-contiguous along N -> coalesced) ----------------------
    const int n = n_base + j;
    if (n < N) {
        #pragma unroll
        for (int k = 0; k < 8; ++k) {
            int m = m_base + m0 + 4 * k;
            if (m < M) out[(size_t)m * N + n] = prod[k];
        }
    }
}

extern "C" void kernel_launch(void* const* d_in, const int* in_sizes, int n_in,
                              void* d_out, int out_size, void* d_ws, size_t ws_size,
                              hipStream_t stream) {
    const float* men = (const float*)d_in[0];   // [M, 128]
    const float* en  = (const float*)d_in[1];   // [N, 128]
    float* out = (float*)d_out;                 // [M, N]
    const int M = in_sizes[0] / 128;            // 256
    const int N = in_sizes[1] / 128;            // 20000

    dim3 grid((N + NT - 1) / NT, (M + MT - 1) / MT);
    hipLaunchKernelGGL(IntersectionalVolumeRatio_kernel, grid, dim3(256), 0, stream,
                       men, en, out, M, N);
}